// ImageQuadratic_19722489823912
// MI455X (gfx1250) — compile-verified
//
#include <hip/hip_runtime.h>
#include <math.h>

#define D_  32
#define H_  128
#define W_  128
#define DHW 524288      // 32*128*128
#define T_  8
#define C_  4
#define XT_SIZE   (T_*C_*DHW)        // 16777216
#define GRID_SIZE (T_*DHW*3)         // 12582912
#define SS_OFF    (XT_SIZE + GRID_SIZE)
#define MM_OFF    (SS_OFF + T_)
#define PI_F 3.14159265358979323846f

// workspace layout (floats): [0..239] beta[a][j][t], [256..639] packed WMMA B
#define WS_BPACK_OFF 256

typedef float v2f __attribute__((ext_vector_type(2)));
typedef float v8f __attribute__((ext_vector_type(8)));

__device__ __forceinline__ int clampi(int v, int lo, int hi) {
  return v < lo ? lo : (v > hi ? hi : v);
}

// ---------------------------------------------------------------------------
// Kernel A: build beta (idx-selected), pack WMMA B operands, reg_ss, reg_mm
// ---------------------------------------------------------------------------
__global__ __launch_bounds__(256) void beta_reg_kernel(
    const float* __restrict__ betas0, const float* __restrict__ betas1,
    const float* __restrict__ betas2, const float* __restrict__ positions,
    const float* __restrict__ centers, const int* __restrict__ idx,
    float* __restrict__ ws, float* __restrict__ out)
{
  __shared__ float sB[240];
  __shared__ float red[256];
  const int tid = threadIdx.x;

  // ---- phase 1: beta columns (j,t), rows [tanh(b0); R; tanh(b2)] ----
  if (tid < 24) {
    const int j = tid / 8, t = tid % 8;
    const int ti = idx[t];
    float col[10];
    col[0] = tanhf(betas0[j * 8 + ti]);
    const float a0 = tanhf(betas1[0 * 8 + ti]) * PI_F;
    const float a1 = tanhf(betas1[1 * 8 + ti]) * PI_F;
    const float a2 = tanhf(betas1[2 * 8 + ti]) * PI_F;
    const float c0 = cosf(a0), s0 = sinf(a0);
    const float c1 = cosf(a1), s1 = sinf(a1);
    const float c2 = cosf(a2), s2 = sinf(a2);
    // R = X(a0) @ Y(a1) @ Z(a2); beta rows 1..3 get column j of R
    if (j == 0)      { col[1] = c0*c1;             col[2] = s0*c1;             col[3] = -s1;   }
    else if (j == 1) { col[1] = -s0*c2 + c0*s1*s2; col[2] = c0*c2 + s0*s1*s2;  col[3] = c1*s2; }
    else             { col[1] = s0*s2 + c0*s1*c2;  col[2] = -c0*s2 + s0*s1*c2; col[3] = c1*c2; }
    #pragma unroll
    for (int i = 0; i < 6; ++i) col[4 + i] = tanhf(betas2[i * 24 + j * 8 + ti]);
    #pragma unroll
    for (int a = 0; a < 10; ++a) {
      const float v = col[a];
      sB[a * 24 + j * 8 + t] = v;
      ws[a * 24 + j * 8 + t] = v;
    }
  }
  __syncthreads();

  // ---- phase 1b: pack B tiles in exact per-lane WMMA layout ----
  // dst = ((tile*3 + s)*32 + lane)*2 + r ; value = beta[K][2-(N%3)][N/3]
  // with N = tile*16 + (lane&15), K = s*4 + 2*(lane>>4) + r (0 if K>=10 or N>=24)
  for (int dst = tid; dst < 384; dst += 256) {
    const int r    = dst & 1;
    const int lane = (dst >> 1) & 31;
    const int q    = dst >> 6;          // 0..5
    const int s    = q % 3;
    const int tile = q / 3;
    const int N    = tile * 16 + (lane & 15);
    const int K    = s * 4 + 2 * (lane >> 4) + r;
    float val = 0.f;
    if (K < 10 && N < 24) {
      const int t = N / 3, b = N % 3;
      val = sB[K * 24 + (2 - b) * 8 + t];
    }
    ws[WS_BPACK_OFF + dst] = val;
  }

  // ---- phase 2: reg_ss ----
  float acc[8] = {0.f,0.f,0.f,0.f,0.f,0.f,0.f,0.f};
  const float sc0 = 31.f, sc1 = 127.f, sc2 = 127.f;
  for (int n = tid; n < 200; n += 256) {
    const float q0 = -1.f + 2.f * centers[n * 3 + 0] / sc0;
    const float q1 = -1.f + 2.f * centers[n * 3 + 1] / sc1;
    const float q2 = -1.f + 2.f * centers[n * 3 + 2] / sc2;
    const float Qq[10] = {1.f, q0, q1, q2, q0*q0, q1*q1, q2*q2, q0*q1, q0*q2, q1*q2};
    const float scb[3] = {sc0, sc1, sc2};
    #pragma unroll
    for (int t = 0; t < 8; ++t) {
      const int ti = idx[t];
      float s = 0.f;
      #pragma unroll
      for (int b = 0; b < 3; ++b) {
        float mv = 0.f;
        #pragma unroll
        for (int a = 0; a < 10; ++a) mv += sB[a * 24 + b * 8 + t] * Qq[a];
        const float P = -1.f + 2.f * positions[n * 24 + b * 8 + ti] / scb[b];
        const float d = mv - P;
        s += d * d;
      }
      acc[t] += s;
    }
  }
  for (int t = 0; t < 8; ++t) {
    red[tid] = acc[t];
    __syncthreads();
    for (int off = 128; off > 0; off >>= 1) {
      if (tid < off) red[tid] += red[tid + off];
      __syncthreads();
    }
    if (tid == 0) out[SS_OFF + t] = red[0] * (1.0f / 200.0f);
    __syncthreads();
  }

  // ---- phase 3: reg_mm (det-Jacobian at 4 tessellation centers) ----
  if (tid < 8) {
    const int t = tid;
    const float tp[4][3] = {{8.f,32.f,64.f},{8.f,96.f,64.f},{24.f,32.f,64.f},{24.f,96.f,64.f}};
    float acc2 = 0.f;
    #pragma unroll
    for (int k = 0; k < 4; ++k) {
      const float px = tp[k][0], py = tp[k][1], pz = tp[k][2];
      float J[3][3];
      #pragma unroll
      for (int col = 0; col < 3; ++col) {
        #define SBV(a) sB[(a) * 24 + col * 8 + t]
        J[col][0] = SBV(1) + 2.f*SBV(4)*px + SBV(7)*py + SBV(8)*pz;
        J[col][1] = SBV(2) + 2.f*SBV(5)*py + SBV(7)*px + SBV(9)*pz;
        J[col][2] = SBV(3) + 2.f*SBV(6)*pz + SBV(8)*px + SBV(9)*py;
        #undef SBV
      }
      const float det = J[0][0]*(J[1][1]*J[2][2] - J[1][2]*J[2][1])
                      - J[0][1]*(J[1][0]*J[2][2] - J[1][2]*J[2][0])
                      + J[0][2]*(J[1][0]*J[2][1] - J[1][1]*J[2][0]);
      const float dd = det - 1.f;
      acc2 += dd * dd;
    }
    out[MM_OFF + t] = acc2;
  }
}

// ---------------------------------------------------------------------------
// Kernel B: WMMA basis x beta -> grid, then trilinear grid_sample -> X_t
// One wave32 handles 16 consecutive voxels (along W) for all 8 t-slices.
// ---------------------------------------------------------------------------
__global__ __launch_bounds__(32) void grid_sample_kernel(
    const float* __restrict__ x, const float* __restrict__ ws,
    float* __restrict__ out)
{
  __shared__ float lg[16][26];   // [voxel M][column N] (24 cols + pad)
  const int lane = threadIdx.x;
  const int mm = lane & 15;      // M (voxel) for A/C layouts; N%16 for B/C layouts
  const int hi = lane >> 4;      // lane-half selector
  const int tile = blockIdx.x;

  // B operands: pre-packed per-lane layout, 6 coalesced 8-byte loads
  const v2f* __restrict__ Bp = (const v2f*)(ws + WS_BPACK_OFF);
  v2f B0[3], B1[3];
  #pragma unroll
  for (int s = 0; s < 3; ++s) {
    B0[s] = Bp[(0 + s) * 32 + lane];
    B1[s] = Bp[(3 + s) * 32 + lane];
  }

  // quadratic basis of this lane's voxel (identity flow, normalized [-1,1])
  const int v  = tile * 16 + mm;
  const int wwi = v & 127;
  const int hhi = (v >> 7) & 127;
  const int ddi = v >> 14;
  const float xx = 2.f * (float)ddi * (1.f / 31.f)  - 1.f;
  const float yy = 2.f * (float)hhi * (1.f / 127.f) - 1.f;
  const float zz = 2.f * (float)wwi * (1.f / 127.f) - 1.f;

  // A (16x4 f32): lane = M, K = 4*s + 2*hi + vgpr
  v2f A[3];
  A[0].x = hi ? yy      : 1.f;     A[0].y = hi ? zz      : xx;
  A[1].x = hi ? zz * zz : xx * xx; A[1].y = hi ? xx * yy : yy * yy;
  A[2].x = hi ? 0.f     : xx * zz; A[2].y = hi ? 0.f     : yy * zz;

  v8f c0 = {};
  v8f c1 = {};
  #pragma unroll
  for (int s = 0; s < 3; ++s) {
    c0 = __builtin_amdgcn_wmma_f32_16x16x4_f32(false, A[s], false, B0[s], (short)0, c0, false, false);
    c1 = __builtin_amdgcn_wmma_f32_16x16x4_f32(false, A[s], false, B1[s], (short)0, c1, false, false);
  }

  // D tiles -> LDS: C/D layout is (M = r + 8*hi, N = lane%16 [+16 for tile1])
  #pragma unroll
  for (int r = 0; r < 8; ++r) {
    const int M = r + 8 * hi;
    lg[M][mm] = c0[r];
    if (mm < 8) lg[M][16 + mm] = c1[r];
  }
  __syncthreads();

  // grid output: 384 floats per tile, written as 12 coalesced 32-lane stores
  // element e = t*48 + m*3 + b  ->  out[XT_SIZE + (t*DHW + tile*16)*3 + (e%48)]
  #pragma unroll
  for (int j = 0; j < 12; ++j) {
    const int e   = j * 32 + lane;   // 0..383
    const int t   = e / 48;
    const int rem = e - t * 48;
    const int m   = rem / 3;
    const int b   = rem - m * 3;
    out[XT_SIZE + (t * DHW + tile * 16) * 3 + rem] = lg[m][t * 3 + b];
  }

  // trilinear sampling: half-waves take alternating t-slices
  #pragma unroll
  for (int k = 0; k < 4; ++k) {
    const int t = 2 * k + hi;
    const float gx = lg[mm][t * 3 + 0];   // -> W
    const float gy = lg[mm][t * 3 + 1];   // -> H
    const float gz = lg[mm][t * 3 + 2];   // -> D
    const float ix = ((gx + 1.f) * 128.f - 1.f) * 0.5f;
    const float iy = ((gy + 1.f) * 128.f - 1.f) * 0.5f;
    const float iz = ((gz + 1.f) * 32.f  - 1.f) * 0.5f;
    const float x0f = floorf(ix), y0f = floorf(iy), z0f = floorf(iz);
    const float tx = ix - x0f, ty = iy - y0f, tz = iz - z0f;
    const int x0 = (int)x0f, y0 = (int)y0f, z0 = (int)z0f;

    float acc0 = 0.f, acc1 = 0.f, acc2 = 0.f, acc3 = 0.f;
    const float* __restrict__ xb = x + (size_t)t * C_ * DHW;
    #pragma unroll
    for (int dz = 0; dz < 2; ++dz) {
      #pragma unroll
      for (int dy = 0; dy < 2; ++dy) {
        #pragma unroll
        for (int dx = 0; dx < 2; ++dx) {
          const int xi = x0 + dx, yi = y0 + dy, zi = z0 + dz;
          float wt = (dx ? tx : 1.f - tx) * (dy ? ty : 1.f - ty) * (dz ? tz : 1.f - tz);
          const bool valid = (xi >= 0) & (xi < 128) & (yi >= 0) & (yi < 128) &
                             (zi >= 0) & (zi < 32);
          wt = valid ? wt : 0.f;
          const int lin = (clampi(zi, 0, 31) * 128 + clampi(yi, 0, 127)) * 128 +
                          clampi(xi, 0, 127);
          acc0 += wt * xb[0 * DHW + lin];
          acc1 += wt * xb[1 * DHW + lin];
          acc2 += wt * xb[2 * DHW + lin];
          acc3 += wt * xb[3 * DHW + lin];
        }
      }
    }
    out[(t * C_ + 0) * DHW + v] = acc0;
    out[(t * C_ + 1) * DHW + v] = acc1;
    out[(t * C_ + 2) * DHW + v] = acc2;
    out[(t * C_ + 3) * DHW + v] = acc3;
  }
}

// ---------------------------------------------------------------------------
extern "C" void kernel_launch(void* const* d_in, const int* in_sizes, int n_in,
                              void* d_out, int out_size, void* d_ws, size_t ws_size,
                              hipStream_t stream) {
  const float* x    = (const float*)d_in[0];
  const int*   idx  = (const int*)  d_in[1];
  const float* b0   = (const float*)d_in[2];
  const float* b1   = (const float*)d_in[3];
  const float* b2   = (const float*)d_in[4];
  const float* pos  = (const float*)d_in[5];
  const float* cen  = (const float*)d_in[6];
  float* out = (float*)d_out;
  float* ws  = (float*)d_ws;   // beta (240 f) + packed WMMA B (384 f @ off 256)

  beta_reg_kernel<<<1, 256, 0, stream>>>(b0, b1, b2, pos, cen, idx, ws, out);
  grid_sample_kernel<<<DHW / 16, 32, 0, stream>>>(x, ws, out);
}